// LearnedOuter_34574486733376
// MI455X (gfx1250) — compile-verified
//
#include <hip/hip_runtime.h>

typedef float v2f __attribute__((ext_vector_type(2)));
typedef float v8f __attribute__((ext_vector_type(8)));

#define DIM_IN  256
#define DIM_OUT 128
#define DIM_H   16
#define NB      2
#define NN      512
#define LN_EPS  1e-5f

// ---------------------------------------------------------------------------
// Kernel 1: LayerNorm(256) + projection 256 -> 32, split into a / bb panels.
// One block (256 threads) per (b,n) row.
// ---------------------------------------------------------------------------
__global__ __launch_bounds__(256) void ln_proj_kernel(
    const float* __restrict__ feats,
    const float* __restrict__ gamma,
    const float* __restrict__ beta,
    const float* __restrict__ W_in,
    const float* __restrict__ b_in,
    float* __restrict__ Abuf,   // [B*N][16]
    float* __restrict__ Bbuf)   // [B*N][16]
{
    __shared__ float xsh[DIM_IN];
    __shared__ float red[256];

    const int row = blockIdx.x;     // b*512 + n
    const int tid = threadIdx.x;

    const float v = feats[row * DIM_IN + tid];

    // mean
    red[tid] = v;
    __syncthreads();
    for (int s = 128; s > 0; s >>= 1) {
        if (tid < s) red[tid] += red[tid + s];
        __syncthreads();
    }
    const float mu = red[0] * (1.0f / DIM_IN);
    __syncthreads();

    // variance (biased, matches jnp.var)
    red[tid] = v * v;
    __syncthreads();
    for (int s = 128; s > 0; s >>= 1) {
        if (tid < s) red[tid] += red[tid + s];
        __syncthreads();
    }
    const float var = red[0] * (1.0f / DIM_IN) - mu * mu;
    const float inv = rsqrtf(var + LN_EPS);

    xsh[tid] = (v - mu) * inv * gamma[tid] + beta[tid];
    __syncthreads();

    // projection: 32 output columns, threads 0..31 each do a 256-dot
    if (tid < 2 * DIM_H) {
        float acc = b_in[tid];
#pragma unroll 8
        for (int k = 0; k < DIM_IN; ++k)
            acc += xsh[k] * W_in[k * (2 * DIM_H) + tid];
        if (tid < DIM_H) Abuf[row * DIM_H + tid] = acc;
        else             Bbuf[row * DIM_H + (tid - DIM_H)] = acc;
    }
}

// ---------------------------------------------------------------------------
// Kernel 2: for each (b,j): T[x][o] = sum_y bb[j][y]*Wr[x][y][o] (LDS),
// then out[b,:,j,:] = A[b] (512x16) @ T (16x128) via V_WMMA_F32_16X16X4_F32.
// Block = 256 threads = 8 waves; wave owns 4 i-tiles x 8 o-tiles.
// ---------------------------------------------------------------------------
__global__ __launch_bounds__(256) void outer_wmma_kernel(
    const float* __restrict__ Abuf,
    const float* __restrict__ Bbuf,
    const float* __restrict__ W_out,   // [(16*16)][128]
    const float* __restrict__ b_out,   // [128]
    float* __restrict__ out)           // [B][N][N][128]
{
    __shared__ float Alds[NN * DIM_H];        // 32 KB: A panel for this b
    __shared__ float Tlds[DIM_H * DIM_OUT];   //  8 KB: per-j reduced weight tile
    __shared__ float bbsh[DIM_H];

    const int j    = blockIdx.x;
    const int b    = blockIdx.y;
    const int tid  = threadIdx.x;
    const int lane = tid & 31;
    const int wave = tid >> 5;

    if (tid < DIM_H)
        bbsh[tid] = Bbuf[(b * NN + j) * DIM_H + tid];
    for (int idx = tid; idx < NN * DIM_H; idx += 256)
        Alds[idx] = Abuf[b * NN * DIM_H + idx];
    __syncthreads();

    // T[x][o] = sum_y bb[y] * W_out[(x*16+y)*128 + o]
    for (int idx = tid; idx < DIM_H * DIM_OUT; idx += 256) {
        const int x = idx >> 7;
        const int o = idx & 127;
        float s = 0.0f;
#pragma unroll
        for (int y = 0; y < DIM_H; ++y)
            s += bbsh[y] * W_out[(x * DIM_H + y) * DIM_OUT + o];
        Tlds[idx] = s;
    }
    __syncthreads();

    // WMMA fragment lane decomposition (ISA 7.12.2, 32-bit A 16x4 / B 4x16):
    // lanes 0-15: rows M=0..15, K pair {0,1}; lanes 16-31: rows M=0..15, K pair {2,3}
    const int hf  = lane >> 4;        // lane half
    const int l16 = lane & 15;
    const int kh  = hf * 2;           // K offset within a K=4 step

    for (int t = 0; t < 4; ++t) {
        const int itile = wave * 4 + t;
        const int arow  = itile * 16 + l16;

        // A fragments for K = 0,4,8,12
        v2f afr[4];
#pragma unroll
        for (int kk = 0; kk < 4; ++kk) {
            afr[kk].x = Alds[arow * DIM_H + kk * 4 + kh];
            afr[kk].y = Alds[arow * DIM_H + kk * 4 + kh + 1];
        }

        for (int ot = 0; ot < 8; ++ot) {
            const int col = ot * 16 + l16;
            v8f acc = {0.f, 0.f, 0.f, 0.f, 0.f, 0.f, 0.f, 0.f};
#pragma unroll
            for (int kk = 0; kk < 4; ++kk) {
                v2f bfr;
                bfr.x = Tlds[(kk * 4 + kh) * DIM_OUT + col];
                bfr.y = Tlds[(kk * 4 + kh + 1) * DIM_OUT + col];
                acc = __builtin_amdgcn_wmma_f32_16x16x4_f32(
                    false, afr[kk], false, bfr, (short)0, acc, false, false);
            }
            const float bo = b_out[col];
            // C/D layout: VGPR r -> row M = r (lanes 0-15) or r+8 (lanes 16-31)
#pragma unroll
            for (int r = 0; r < 8; ++r) {
                const int orow = itile * 16 + r + hf * 8;
                out[(((size_t)b * NN + orow) * NN + j) * DIM_OUT + col] = acc[r] + bo;
            }
        }
    }
}

extern "C" void kernel_launch(void* const* d_in, const int* in_sizes, int n_in,
                              void* d_out, int out_size, void* d_ws, size_t ws_size,
                              hipStream_t stream) {
    (void)in_sizes; (void)n_in; (void)out_size; (void)ws_size;

    const float* feats = (const float*)d_in[0];
    const float* gamma = (const float*)d_in[1];
    const float* beta  = (const float*)d_in[2];
    const float* W_in  = (const float*)d_in[3];
    const float* b_in  = (const float*)d_in[4];
    const float* W_out = (const float*)d_in[5];
    const float* b_out = (const float*)d_in[6];
    float* out = (float*)d_out;

    float* Abuf = (float*)d_ws;                 // 64 KB
    float* Bbuf = Abuf + (size_t)NB * NN * DIM_H; // 64 KB

    ln_proj_kernel<<<dim3(NB * NN), 256, 0, stream>>>(
        feats, gamma, beta, W_in, b_in, Abuf, Bbuf);
    outer_wmma_kernel<<<dim3(NN, NB), 256, 0, stream>>>(
        Abuf, Bbuf, W_out, b_out, out);
}